// GraphConvBlock_79001628443385
// MI455X (gfx1250) — compile-verified
//
#include <hip/hip_runtime.h>

#define NUM_NODES 50000
#define NUM_EDGES 600000
#define DIM 128        // D
#define FEAT 128       // FEATURES
#define KDIM 256       // 2*D
#define MSUB 5         // M-tiles per wave strip (3125 = 625 * 5, exact)

typedef __attribute__((ext_vector_type(2))) float v2f;
typedef __attribute__((ext_vector_type(8))) float v8f;

// ---------------------------------------------------------------------------
// Phase 0: zero the aggregation workspace (poisoned 0xAA by harness).
// ---------------------------------------------------------------------------
__global__ __launch_bounds__(256) void gcb_zero_kernel(float4* __restrict__ agg, int n4) {
    int i = blockIdx.x * blockDim.x + threadIdx.x;
    if (i < n4) agg[i] = make_float4(0.f, 0.f, 0.f, 0.f);
}

// ---------------------------------------------------------------------------
// Phase 1: edge gather + scatter-add.
// One wave (32 lanes) per edge; each lane moves a float4 (32*4 = 128 = DIM).
// agg (25.6 MB) is L2-resident -> f32 atomics resolve in L2.
// ---------------------------------------------------------------------------
__global__ __launch_bounds__(256) void gcb_scatter_kernel(
    const float* __restrict__ node_x,
    const int*   __restrict__ sources,
    const int*   __restrict__ targets,
    float*       __restrict__ agg)
{
    int idx = blockIdx.x * blockDim.x + threadIdx.x;
    int e = idx >> 5;          // edge id
    int c = idx & 31;          // float4 chunk within the 128-wide row
    if (e >= NUM_EDGES) return;

    int src = sources[e];
    int tgt = targets[e];

    const float4 v = *(const float4*)(node_x + (size_t)src * DIM + c * 4);
    float* dst = agg + (size_t)tgt * DIM + c * 4;
    atomicAdd(dst + 0, v.x);
    atomicAdd(dst + 1, v.y);
    atomicAdd(dst + 2, v.z);
    atomicAdd(dst + 3, v.w);
}

// ---------------------------------------------------------------------------
// Phase 2: out = concat(agg, node_x) @ W + b using V_WMMA_F32_16X16X4_F32.
// One wave computes a strip of MSUB=5 16x16 tiles in the M direction sharing
// one 16-wide N tile: per K-step the B fragment (2 b32 loads) is reused by 5
// WMMAs, and the 5 A fragments are b64 loads off a single base address with
// compile-time byte offsets (s * 16 * DIM * 4). 1.4 loads per WMMA.
//
// Fragment layouts (ISA 7.12.2, 32-bit data, wave32):
//   A 16x4 : lanes 0-15 -> {K0,K1}, lanes 16-31 -> {K2,K3}; M = lane%16
//   B 4x16 : lane group hi holds rows K=2*hi and K=2*hi+1; N = lane%16
//   C 16x16: VGPR j -> M = j + 8*hi, N = lane%16
// EXEC is all-1s: 625 strips * 8 n-tiles = 5000 waves exactly, no divergence.
// ---------------------------------------------------------------------------
__global__ __launch_bounds__(128) void gcb_wmma_gemm_kernel(
    const float* __restrict__ agg,     // [NUM_NODES, DIM]   (K = 0..127)
    const float* __restrict__ node_x,  // [NUM_NODES, DIM]   (K = 128..255)
    const float* __restrict__ W,       // [KDIM, FEAT]
    const float* __restrict__ bias,    // [FEAT]
    float*       __restrict__ out)     // [NUM_NODES, FEAT]
{
    const int wave = threadIdx.x >> 5;              // 0..3
    const int lane = threadIdx.x & 31;
    const int wid  = blockIdx.x * 4 + wave;         // 0..4999
    const int strip = wid >> 3;                     // 0..624
    const int nt    = wid & 7;                      // 0..7
    const int row0  = strip * (MSUB * 16);          // 80-row strip
    const int col0  = nt * 16;
    const int lm    = lane & 15;                    // M (A/C) / N (B/C)
    const int hi    = lane >> 4;                    // lane group

    v8f acc[MSUB];
#pragma unroll
    for (int s = 0; s < MSUB; ++s) acc[s] = (v8f){};

    // Single base per concat half; s-th tile at constant byte offset s*16*DIM*4.
    const float* arow0 = agg    + (size_t)(row0 + lm) * DIM;  // K 0..127
    const float* arow1 = node_x + (size_t)(row0 + lm) * DIM;  // K 128..255
    const float* wcol  = W + col0 + lm;                       // stride FEAT per K row

    // K = 0..127 : aggregated half of the concat
#pragma unroll
    for (int k0 = 0; k0 < DIM; k0 += 4) {
        const int ka = k0 + 2 * hi;
        v2f b; b.x = wcol[(size_t)ka * FEAT];
               b.y = wcol[(size_t)(ka + 1) * FEAT];
#pragma unroll
        for (int s = 0; s < MSUB; ++s) {
            v2f a; a.x = arow0[s * 16 * DIM + ka];
                   a.y = arow0[s * 16 * DIM + ka + 1];
            acc[s] = __builtin_amdgcn_wmma_f32_16x16x4_f32(
                        false, a, false, b, (short)0, acc[s], false, false);
        }
    }
    // K = 128..255 : node_x half of the concat
#pragma unroll
    for (int k0 = 0; k0 < DIM; k0 += 4) {
        const int ka = k0 + 2 * hi;
        v2f b; b.x = wcol[(size_t)(DIM + ka) * FEAT];
               b.y = wcol[(size_t)(DIM + ka + 1) * FEAT];
#pragma unroll
        for (int s = 0; s < MSUB; ++s) {
            v2f a; a.x = arow1[s * 16 * DIM + ka];
                   a.y = arow1[s * 16 * DIM + ka + 1];
            acc[s] = __builtin_amdgcn_wmma_f32_16x16x4_f32(
                        false, a, false, b, (short)0, acc[s], false, false);
        }
    }

    const float bv = bias[col0 + lm];
#pragma unroll
    for (int s = 0; s < MSUB; ++s) {
#pragma unroll
        for (int j = 0; j < 8; ++j) {
            const int r = row0 + s * 16 + j + 8 * hi;
            out[(size_t)r * FEAT + col0 + lm] = acc[s][j] + bv;
        }
    }
}

// ---------------------------------------------------------------------------
// Inputs (setup_inputs order): node_x[f32], sources[int], targets[int],
// W[f32 256x128], b[f32 128]. Output: [50000,128] f32.
// Workspace: agg = NUM_NODES*DIM f32 = 25.6 MB.
// ---------------------------------------------------------------------------
extern "C" void kernel_launch(void* const* d_in, const int* in_sizes, int n_in,
                              void* d_out, int out_size, void* d_ws, size_t ws_size,
                              hipStream_t stream) {
    const float* node_x  = (const float*)d_in[0];
    const int*   sources = (const int*)  d_in[1];
    const int*   targets = (const int*)  d_in[2];
    const float* W       = (const float*)d_in[3];
    const float* bias    = (const float*)d_in[4];
    float*       out     = (float*)d_out;
    float*       agg     = (float*)d_ws;

    // Phase 0: zero agg
    const int n4 = NUM_NODES * DIM / 4;   // 1.6M float4
    gcb_zero_kernel<<<(n4 + 255) / 256, 256, 0, stream>>>((float4*)agg, n4);

    // Phase 1: scatter-add (one wave per edge)
    const long long threads = (long long)NUM_EDGES * 32;   // 19.2M
    gcb_scatter_kernel<<<(int)((threads + 255) / 256), 256, 0, stream>>>(
        node_x, sources, targets, agg);

    // Phase 2: WMMA GEMM; 5000 strip-waves, 4 waves/block -> 1250 blocks
    gcb_wmma_gemm_kernel<<<1250, 128, 0, stream>>>(agg, node_x, W, bias, out);
}